// MultiheadAttention_78460462563929
// MI455X (gfx1250) — compile-verified
//
#include <hip/hip_runtime.h>
#include <hip/hip_bf16.h>

// ---------------------------------------------------------------------------
// Multi-head attention forward, MI455X (gfx1250), fp32 WMMA path.
//   B=2, T=2048, D=768, H=12, Dh=64
// Outputs (concatenated in d_out): o [B,T,D] fp32, attention [B,H,T,T] fp32.
//
// Pipeline:
//  1) QKV GEMM (fp32 WMMA 16x16x4, async-to-LDS staged tiles), epilogue
//     scatters into [B,H,T,Dh] Q/K/V.
//  2) Fused attention per (b,h,16-query block): S strip [16x2048] in LDS,
//     softmax in LDS, P written once to HBM, P*V via WMMA from LDS.
//  3) O-projection GEMM.
// ---------------------------------------------------------------------------

typedef float v2f __attribute__((ext_vector_type(2)));
typedef float v8f __attribute__((ext_vector_type(8)));
typedef int   v4i __attribute__((ext_vector_type(4)));

static constexpr int Bb   = 2;
static constexpr int Tt   = 2048;
static constexpr int Dd   = 768;
static constexpr int Hh   = 12;
static constexpr int Dh   = 64;
static constexpr int M_   = Bb * Tt;      // 4096 rows
static constexpr int N3   = 3 * Dd;       // 2304

#if __has_builtin(__builtin_amdgcn_global_load_async_to_lds_b128) && \
    __has_builtin(__builtin_amdgcn_s_wait_asynccnt)
#define ASYNC_LDS 1
#else
#define ASYNC_LDS 0
#warning "gfx1250 async-to-LDS builtins unavailable; using synchronous staging"
#endif

// Copy 16 contiguous bytes global -> LDS (per-lane), using the CDNA5 async
// path (GLOBAL_LOAD_ASYNC_TO_LDS_B128, ASYNCcnt-tracked) when available.
__device__ __forceinline__ void stage16(const float* __restrict__ g, float* l) {
#if ASYNC_LDS
    __builtin_amdgcn_global_load_async_to_lds_b128(
        (__attribute__((address_space(1))) v4i*)g,
        (__attribute__((address_space(3))) v4i*)l,
        /*offset=*/0, /*cpol=*/0);
#else
    *(float4*)l = *(const float4*)g;
#endif
}

__device__ __forceinline__ void stage_wait() {
#if ASYNC_LDS
    __builtin_amdgcn_s_wait_asynccnt(0);
#endif
    __syncthreads();
}

// D = A(16x4) * B(4x16) + C, fp32 matrix core.
// Fragment layout (ISA 7.12.2, 32-bit A 16x4): vgpr p, lanes[0..15] hold
// K=p, lanes[16..31] hold K=p+2; M (A) / N (B) = lane%16.
// C/D: vgpr r, lanes[0..15] -> M=r, lanes[16..31] -> M=r+8; N = lane%16.
__device__ __forceinline__ v8f wmma4(v2f a, v2f b, v8f c) {
    return __builtin_amdgcn_wmma_f32_16x16x4_f32(
        /*neg_a=*/false, a, /*neg_b=*/false, b,
        /*c_mod=*/(short)0, c, /*reuse_a=*/false, /*reuse_b=*/false);
}

// ---------------------------------------------------------------------------
// Kernel 1: fused QKV projection. X[4096,768] x W[768,2304] + bias,
// epilogue scatters into Q/K/V [B,H,T,Dh] (fuses reshape+transpose).
// Block = 256 threads (8 waves), tile 64x64, BK=16.
// ---------------------------------------------------------------------------
__global__ __launch_bounds__(256) void qkv_gemm_kernel(
    const float* __restrict__ X, const float* __restrict__ W,
    const float* __restrict__ bias,
    float* __restrict__ Q, float* __restrict__ K, float* __restrict__ V)
{
    __shared__ float As[64 * 16];
    __shared__ float Bs[16 * 64];

    const int tid  = threadIdx.x;
    const int row0 = blockIdx.x * 64;   // over M=4096 (64 blocks)
    const int col0 = blockIdx.y * 64;   // over N=2304 (36 blocks)

    const int w    = tid >> 5;
    const int lane = tid & 31;
    const int half = lane >> 4;
    const int lm   = lane & 15;
    const int wm   = (w & 3) * 16;      // M sub-tile within block
    const int wn   = (w >> 2) * 32;     // N range: two 16-wide tiles

    v8f c0 = {}; v8f c1 = {};

    const int ra = tid >> 2, ca = (tid & 3) * 4;    // A-stage coords
    const int rb = tid >> 4, cb = (tid & 15) * 4;   // B-stage coords

    for (int k0 = 0; k0 < Dd; k0 += 16) {
        // stage A tile 64x16 and B tile 16x64 directly into LDS (async DMA)
        stage16(X + (size_t)(row0 + ra) * Dd + k0 + ca, As + ra * 16 + ca);
        stage16(W + (size_t)(k0 + rb) * N3 + col0 + cb, Bs + rb * 64 + cb);
        stage_wait();
#pragma unroll
        for (int kk = 0; kk < 16; kk += 4) {
            const int kc = kk + 2 * half;
            v2f a;  a.x  = As[(wm + lm) * 16 + kc];
                    a.y  = As[(wm + lm) * 16 + kc + 1];
            v2f b0; b0.x = Bs[kc * 64 + wn + lm];
                    b0.y = Bs[(kc + 1) * 64 + wn + lm];
            v2f b1; b1.x = Bs[kc * 64 + wn + 16 + lm];
                    b1.y = Bs[(kc + 1) * 64 + wn + 16 + lm];
            c0 = wmma4(a, b0, c0);
            c1 = wmma4(a, b1, c1);
        }
        __syncthreads();
    }

    // Epilogue: scatter e -> (part, head, dh); row -> (b, t).
#pragma unroll
    for (int r = 0; r < 8; ++r) {
        const int gm = row0 + wm + r + 8 * half;
        const int b_ = gm >> 11;          // /2048
        const int t_ = gm & 2047;
#pragma unroll
        for (int nt = 0; nt < 2; ++nt) {
            const int e   = col0 + wn + nt * 16 + lm;
            const float v = (nt == 0 ? c0[r] : c1[r]) + bias[e];
            const int part = e / Dd;
            const int rem  = e - part * Dd;
            const int h    = rem >> 6;
            const int dh   = rem & 63;
            float* dst = (part == 0) ? Q : (part == 1 ? K : V);
            dst[(((size_t)b_ * Hh + h) * Tt + t_) * Dh + dh] = v;
        }
    }
}

// ---------------------------------------------------------------------------
// Kernel 2: fused attention for one (b, h, 16-query block).
// S strip [16 x 2048] lives in LDS (128 KB; CDNA5 has 320 KB per WGP).
// ---------------------------------------------------------------------------
__global__ __launch_bounds__(256) void attn_fused_kernel(
    const float* __restrict__ Q, const float* __restrict__ K,
    const float* __restrict__ V,
    float* __restrict__ att, float* __restrict__ vals)
{
    extern __shared__ float smem[];
    float* Sbuf = smem;                  // 16*2048
    float* Qs   = Sbuf + 16 * 2048;      // 16*64
    float* red  = Qs + 16 * 64;          // 16*16
    float* rowm = red + 256;             // 16
    float* rows = rowm + 16;             // 16
    float* pbuf = rows + 16;             // 8*256

    const int tid = threadIdx.x;
    const int qb  = blockIdx.x & 127;
    const int h   = (blockIdx.x >> 7) % Hh;
    const int b   = blockIdx.x / (128 * Hh);
    const int q0  = qb * 16;

    const size_t base = ((size_t)b * Hh + h) * (size_t)Tt * Dh;
    const float* qp = Q + base + (size_t)q0 * Dh;
    const float* kp = K + base;
    const float* vp = V + base;

    // Q tile [16x64]: pure async copy (scale folded into softmax exp)
    {
        const int r = tid >> 4, c = (tid & 15) * 4;
        stage16(qp + r * Dh + c, Qs + r * 64 + c);
    }
    stage_wait();

    const int w = tid >> 5, lane = tid & 31;
    const int half = lane >> 4, lm = lane & 15;

    // ---- phase 1: S = Q K^T (unscaled) ----
    for (int i = 0; i < 16; ++i) {
        const int j = w * 16 + i;        // key tile 0..127
        if (i + 1 < 16)                  // prefetch next key tile row
            __builtin_prefetch(kp + (size_t)((j + 1) * 16 + lm) * Dh, 0, 0);
        v8f c = {};
#pragma unroll
        for (int kk = 0; kk < Dh; kk += 4) {
            const int kc = kk + 2 * half;
            v2f a;  a.x = Qs[lm * 64 + kc];
                    a.y = Qs[lm * 64 + kc + 1];
            v2f bf; bf.x = kp[(size_t)(j * 16 + lm) * Dh + kc];
                    bf.y = kp[(size_t)(j * 16 + lm) * Dh + kc + 1];
            c = wmma4(a, bf, c);
        }
#pragma unroll
        for (int r = 0; r < 8; ++r)
            Sbuf[(r + 8 * half) * 2048 + j * 16 + lm] = c[r];
    }
    __syncthreads();

    // ---- phase 2: softmax over 2048 cols, 16 threads per row ----
    // softmax(scale*S): max over raw S, exp((S-m)*scale).
    const float scale = 0.125f;          // 1/sqrt(64)
    const int srow = tid >> 4, sseg = tid & 15;
    float* Srow = Sbuf + srow * 2048 + sseg * 128;

    float mx = -3.402823e38f;
    for (int c = 0; c < 128; ++c) mx = fmaxf(mx, Srow[c]);
    red[srow * 16 + sseg] = mx;
    __syncthreads();
    if (sseg == 0) {
        float m = red[srow * 16];
        for (int i = 1; i < 16; ++i) m = fmaxf(m, red[srow * 16 + i]);
        rowm[srow] = m;
    }
    __syncthreads();
    const float m = rowm[srow];
    float sum = 0.0f;
    for (int c = 0; c < 128; ++c) {
        const float e = __expf((Srow[c] - m) * scale);
        Srow[c] = e;
        sum += e;
    }
    red[srow * 16 + sseg] = sum;
    __syncthreads();
    if (sseg == 0) {
        float s = 0.0f;
        for (int i = 0; i < 16; ++i) s += red[srow * 16 + i];
        rows[srow] = s;
    }
    __syncthreads();
    const float inv = 1.0f / rows[srow];
    float* attRow = att + (((size_t)b * Hh + h) * Tt + (q0 + srow)) * (size_t)Tt
                        + sseg * 128;
    for (int c = 0; c < 128; c += 4) {
        float4 p;
        p.x = Srow[c]     * inv;
        p.y = Srow[c + 1] * inv;
        p.z = Srow[c + 2] * inv;
        p.w = Srow[c + 3] * inv;
        Srow[c] = p.x; Srow[c + 1] = p.y; Srow[c + 2] = p.z; Srow[c + 3] = p.w;
        *(float4*)(attRow + c) = p;      // attention written exactly once
    }
    __syncthreads();

    // ---- phase 3: out = P x V ----
    const int ntile = (w & 3) * 16;      // N sub-tile of 64
    const int khalf = w >> 2;            // K half: 0..1023 / 1024..2047
    v8f acc = {};
    for (int chunk = 0; chunk < 64; ++chunk) {
        const int kbase = khalf * 1024 + chunk * 16;
#pragma unroll
        for (int kk = 0; kk < 16; kk += 4) {
            const int kc = kbase + kk + 2 * half;
            v2f a;  a.x = Sbuf[lm * 2048 + kc];
                    a.y = Sbuf[lm * 2048 + kc + 1];
            v2f bf; bf.x = vp[(size_t)kc * Dh + ntile + lm];
                    bf.y = vp[(size_t)(kc + 1) * Dh + ntile + lm];
            acc = wmma4(a, bf, acc);
        }
    }
#pragma unroll
    for (int r = 0; r < 8; ++r)
        pbuf[w * 256 + (r + 8 * half) * 16 + lm] = acc[r];
    __syncthreads();

    // reduce the two K-half partials, write values in [B,T,D] layout
    for (int idx = tid; idx < 16 * 64; idx += 256) {
        const int row = idx >> 6, col = idx & 63;
        const int nt = col >> 4, lc = col & 15;
        const float v = pbuf[nt * 256 + row * 16 + lc]
                      + pbuf[(nt + 4) * 256 + row * 16 + lc];
        vals[((size_t)b * Tt + q0 + row) * Dd + h * Dh + col] = v;
    }
}

// ---------------------------------------------------------------------------
// Kernel 3: output projection. vals[4096,768] x Wo[768,768] + bias -> o.
// ---------------------------------------------------------------------------
__global__ __launch_bounds__(256) void o_gemm_kernel(
    const float* __restrict__ X, const float* __restrict__ W,
    const float* __restrict__ bias, float* __restrict__ out)
{
    __shared__ float As[64 * 16];
    __shared__ float Bs[16 * 64];

    const int tid  = threadIdx.x;
    const int row0 = blockIdx.x * 64;   // M=4096 -> 64 blocks
    const int col0 = blockIdx.y * 64;   // N=768  -> 12 blocks

    const int w    = tid >> 5;
    const int lane = tid & 31;
    const int half = lane >> 4;
    const int lm   = lane & 15;
    const int wm   = (w & 3) * 16;
    const int wn   = (w >> 2) * 32;

    v8f c0 = {}; v8f c1 = {};

    const int ra = tid >> 2, ca = (tid & 3) * 4;
    const int rb = tid >> 4, cb = (tid & 15) * 4;

    for (int k0 = 0; k0 < Dd; k0 += 16) {
        stage16(X + (size_t)(row0 + ra) * Dd + k0 + ca, As + ra * 16 + ca);
        stage16(W + (size_t)(k0 + rb) * Dd + col0 + cb, Bs + rb * 64 + cb);
        stage_wait();
#pragma unroll
        for (int kk = 0; kk < 16; kk += 4) {
            const int kc = kk + 2 * half;
            v2f a;  a.x  = As[(wm + lm) * 16 + kc];
                    a.y  = As[(wm + lm) * 16 + kc + 1];
            v2f b0; b0.x = Bs[kc * 64 + wn + lm];
                    b0.y = Bs[(kc + 1) * 64 + wn + lm];
            v2f b1; b1.x = Bs[kc * 64 + wn + 16 + lm];
                    b1.y = Bs[(kc + 1) * 64 + wn + 16 + lm];
            c0 = wmma4(a, b0, c0);
            c1 = wmma4(a, b1, c1);
        }
        __syncthreads();
    }

#pragma unroll
    for (int r = 0; r < 8; ++r) {
        const int gm = row0 + wm + r + 8 * half;
#pragma unroll
        for (int nt = 0; nt < 2; ++nt) {
            const int e = col0 + wn + nt * 16 + lm;
            out[(size_t)gm * Dd + e] = (nt == 0 ? c0[r] : c1[r]) + bias[e];
        }
    }
}

// ---------------------------------------------------------------------------
extern "C" void kernel_launch(void* const* d_in, const int* in_sizes, int n_in,
                              void* d_out, int out_size, void* d_ws, size_t ws_size,
                              hipStream_t stream)
{
    const float* x      = (const float*)d_in[0];
    const float* qkv_w  = (const float*)d_in[1];
    const float* qkv_b  = (const float*)d_in[2];
    const float* o_w    = (const float*)d_in[3];
    const float* o_b    = (const float*)d_in[4];

    float* out     = (float*)d_out;
    float* o_out   = out;                                    // [B,T,D]
    float* att_out = out + (size_t)Bb * Tt * Dd;             // [B,H,T,T]

    float* ws = (float*)d_ws;
    const size_t QSZ = (size_t)Bb * Hh * Tt * Dh;            // 3.1M floats each
    float* Q    = ws;
    float* K    = ws + QSZ;
    float* V    = ws + 2 * QSZ;
    float* vals = ws + 3 * QSZ;                              // [B,T,D]

    const int attn_smem = (16 * 2048 + 16 * 64 + 256 + 16 + 16 + 8 * 256)
                          * (int)sizeof(float);              // 144,512 B
    (void)hipFuncSetAttribute((const void*)attn_fused_kernel,
                              hipFuncAttributeMaxDynamicSharedMemorySize,
                              attn_smem);

    dim3 g1(M_ / 64, N3 / 64);                               // 64 x 36
    qkv_gemm_kernel<<<g1, 256, 0, stream>>>(x, qkv_w, qkv_b, Q, K, V);

    const int nblk = Bb * Hh * (Tt / 16);                    // 3072
    attn_fused_kernel<<<nblk, 256, attn_smem, stream>>>(Q, K, V, att_out, vals);

    dim3 g3(M_ / 64, Dd / 64);                               // 64 x 12
    o_gemm_kernel<<<g3, 256, 0, stream>>>(vals, o_w, o_b, o_out);
}